// GraphSAGE_9113920602386
// MI455X (gfx1250) — compile-verified
//
#include <hip/hip_runtime.h>

typedef __attribute__((ext_vector_type(2))) float v2f;
typedef __attribute__((ext_vector_type(8))) float v8f;

#define F_IN 128

// ---------------------------------------------------------------- zero fill
__global__ void zero_kernel(float* __restrict__ p, long long n) {
  long long i = (long long)blockIdx.x * blockDim.x + threadIdx.x;
  long long stride = (long long)gridDim.x * blockDim.x;
  for (; i < n; i += stride) p[i] = 0.0f;
}

// ---------------------------------------------------------------- degree
__global__ void deg_kernel(const int* __restrict__ dst, float* __restrict__ deg, int E) {
  int e = blockIdx.x * blockDim.x + threadIdx.x;
  if (e < E) atomicAdd(&deg[dst[e]], 1.0f);
}

// ------------------------------------------------- edge gather + scatter-add
// One wave per edge: 32 lanes x float4 = the full 128-float feature row.
// Gather is a fully coalesced 512B read; atomics hit 512B of consecutive
// addresses so L2 can combine them.
__global__ void scatter_kernel(const float* __restrict__ feat,
                               const int* __restrict__ src,
                               const int* __restrict__ dst,
                               float* __restrict__ agg, int E) {
  long long gid = (long long)blockIdx.x * blockDim.x + threadIdx.x;
  int e = (int)(gid >> 5);
  if (e >= E) return;
  int chunk = (int)(gid & 31);
  int s = src[e];
  int d = dst[e];
  const float4 v = *reinterpret_cast<const float4*>(feat + (size_t)s * F_IN + chunk * 4);
  float* base = agg + (size_t)d * F_IN + chunk * 4;
  atomicAdd(base + 0, v.x);
  atomicAdd(base + 1, v.y);
  atomicAdd(base + 2, v.z);
  atomicAdd(base + 3, v.w);
}

// ------------------------------------------------- fused SAGE linear (WMMA)
// out[m][n] = (agg[m]/deg[m]) @ Wl[n].T + bl[n] + xin[m] @ Wr[n].T  (+ReLU)
// One wave computes a 16x32 output tile via V_WMMA_F32_16X16X4_F32,
// reusing the A operand across two 16-wide N tiles.
//
// A operand layout (16x4 f32): lane L holds row M=L&15; VGPR j holds
// K = 2*(L>>4) + j.  B operand (4x16) is symmetric with N=L&15.
// C/D: VGPR j -> M = j + 8*(L>>4), N = L&15.
template <int NOUT, bool RELU>
__global__ void sage_gemm_kernel(const float* __restrict__ agg,
                                 const float* __restrict__ deg,
                                 const float* __restrict__ xin,
                                 const float* __restrict__ Wl,
                                 const float* __restrict__ bl,
                                 const float* __restrict__ Wr,
                                 float* __restrict__ out, int nRows) {
  const int lane = threadIdx.x & 31;
  const int tile = blockIdx.x * (blockDim.x >> 5) + (threadIdx.x >> 5);
  const int nTilesN = NOUT / 32;  // pairs of 16-wide N tiles per M tile
  const int nTilesM = nRows / 16; // N = 50000 is divisible by 16
  if (tile >= nTilesM * nTilesN) return;
  const int tm = tile / nTilesN;
  const int tn = tile % nTilesN;

  const int l15  = lane & 15;
  const int koff = (lane >> 4) * 2;           // 0 or 2
  const int mrow = tm * 16 + l15;             // row this lane serves for A
  const int col0 = tn * 32 + l15;             // first B column
  const int col1 = col0 + 16;                 // second B column

  const float invdeg = 1.0f / fmaxf(deg[mrow], 1.0f);

  v8f acc0 = {};
  v8f acc1 = {};

  // ---- phase 1: (agg/deg) @ Wl.T
  {
    const float* A  = agg + (size_t)mrow * F_IN;
    const float* B0 = Wl  + (size_t)col0 * F_IN;
    const float* B1 = Wl  + (size_t)col1 * F_IN;
#pragma unroll
    for (int k = 0; k < F_IN; k += 4) {
      v2f a = *reinterpret_cast<const v2f*>(A + k + koff);
      a.x *= invdeg;
      a.y *= invdeg;
      v2f b0 = *reinterpret_cast<const v2f*>(B0 + k + koff);
      v2f b1 = *reinterpret_cast<const v2f*>(B1 + k + koff);
      acc0 = __builtin_amdgcn_wmma_f32_16x16x4_f32(false, a, false, b0,
                                                   (short)0, acc0, false, false);
      acc1 = __builtin_amdgcn_wmma_f32_16x16x4_f32(false, a, false, b1,
                                                   (short)0, acc1, false, false);
    }
  }
  // ---- phase 2: x @ Wr.T
  {
    const float* A  = xin + (size_t)mrow * F_IN;
    const float* B0 = Wr  + (size_t)col0 * F_IN;
    const float* B1 = Wr  + (size_t)col1 * F_IN;
#pragma unroll
    for (int k = 0; k < F_IN; k += 4) {
      v2f a  = *reinterpret_cast<const v2f*>(A + k + koff);
      v2f b0 = *reinterpret_cast<const v2f*>(B0 + k + koff);
      v2f b1 = *reinterpret_cast<const v2f*>(B1 + k + koff);
      acc0 = __builtin_amdgcn_wmma_f32_16x16x4_f32(false, a, false, b0,
                                                   (short)0, acc0, false, false);
      acc1 = __builtin_amdgcn_wmma_f32_16x16x4_f32(false, a, false, b1,
                                                   (short)0, acc1, false, false);
    }
  }

  const float bias0 = bl[tn * 32 + l15];
  const float bias1 = bl[tn * 32 + 16 + l15];
  const int mbase = tm * 16 + (lane >> 4) * 8;
#pragma unroll
  for (int j = 0; j < 8; ++j) {
    float v0 = acc0[j] + bias0;
    float v1 = acc1[j] + bias1;
    if (RELU) {
      v0 = fmaxf(v0, 0.0f);
      v1 = fmaxf(v1, 0.0f);
    }
    out[(size_t)(mbase + j) * NOUT + col0] = v0;
    out[(size_t)(mbase + j) * NOUT + col1] = v1;
  }
}

// ---------------------------------------------------------------- log_softmax
// 64 columns per row: one wave32 per row, 2 values per lane, shuffle reduce.
__global__ void log_softmax64_kernel(float* __restrict__ data, int nRows) {
  const int lane = threadIdx.x & 31;
  const int row = blockIdx.x * (blockDim.x >> 5) + (threadIdx.x >> 5);
  if (row >= nRows) return;
  float* p = data + (size_t)row * 64 + lane * 2;
  v2f v = *reinterpret_cast<const v2f*>(p);

  float m = fmaxf(v.x, v.y);
#pragma unroll
  for (int off = 16; off > 0; off >>= 1) m = fmaxf(m, __shfl_xor(m, off, 32));

  float s = expf(v.x - m) + expf(v.y - m);
#pragma unroll
  for (int off = 16; off > 0; off >>= 1) s += __shfl_xor(s, off, 32);

  const float lse = m + logf(s);
  v.x -= lse;
  v.y -= lse;
  *reinterpret_cast<v2f*>(p) = v;
}

// ---------------------------------------------------------------- launcher
extern "C" void kernel_launch(void* const* d_in, const int* in_sizes, int n_in,
                              void* d_out, int out_size, void* d_ws, size_t ws_size,
                              hipStream_t stream) {
  const float* x   = (const float*)d_in[0];
  const int* edge  = (const int*)d_in[1];
  const float* W1l = (const float*)d_in[2];
  const float* b1  = (const float*)d_in[3];
  const float* W1r = (const float*)d_in[4];
  const float* W2l = (const float*)d_in[5];
  const float* b2  = (const float*)d_in[6];
  const float* W2r = (const float*)d_in[7];

  const int N = in_sizes[0] / F_IN;  // 50000
  const int E = in_sizes[1] / 2;     // 640000
  const int* src = edge;
  const int* dst = edge + E;

  // workspace: deg[N] | agg[N*128] (shared by both layers) | h[N*128]
  float* deg = (float*)d_ws;
  float* agg = deg + (size_t)N;
  float* h   = agg + (size_t)N * F_IN;
  float* logits = (float*)d_out;     // layer-2 logits written straight to d_out

  const int scatterBlocks = (int)(((long long)E * 32 + 255) / 256);

  // ---- layer 1
  zero_kernel<<<2048, 256, 0, stream>>>(deg, (long long)N * (F_IN + 1));  // deg+agg
  deg_kernel<<<(E + 255) / 256, 256, 0, stream>>>(dst, deg, E);
  scatter_kernel<<<scatterBlocks, 256, 0, stream>>>(x, src, dst, agg, E);
  {
    const int waves = (N / 16) * (F_IN / 32);
    sage_gemm_kernel<F_IN, true>
        <<<(waves + 7) / 8, 256, 0, stream>>>(agg, deg, x, W1l, b1, W1r, h, N);
  }

  // ---- layer 2
  zero_kernel<<<2048, 256, 0, stream>>>(agg, (long long)N * F_IN);
  scatter_kernel<<<scatterBlocks, 256, 0, stream>>>(h, src, dst, agg, E);
  {
    const int waves = (N / 16) * (64 / 32);
    sage_gemm_kernel<64, false>
        <<<(waves + 7) / 8, 256, 0, stream>>>(agg, deg, h, W2l, b2, W2r, logits, N);
  }

  // ---- log_softmax over 64 columns, in place on d_out
  log_softmax64_kernel<<<(N + 7) / 8, 256, 0, stream>>>(logits, N);
}